// ElmoNer_28071906247321
// MI455X (gfx1250) — compile-verified
//
#include <hip/hip_runtime.h>

typedef __attribute__((ext_vector_type(2))) float v2f;
typedef __attribute__((ext_vector_type(8))) float v8f;

#define B_N 64
#define T_N 512
#define U_N 1024
#define H_N 512
#define K_N 9
#define G_N 1536   // 3*H
#define KP  516    // LDS row stride (512 + 4 pad) -> conflict-free ds_load_b64

// GRU scan partitioning: 32 blocks per direction, each owns 16 h-columns.
#define GRU_THREADS 128
#define NBLK_DIR    32
#define CHUNK       16
#define ROWS        48   // 3 gates * 16 w_hh rows staged in LDS

static __device__ __forceinline__ v8f wmma4(v2f a, v2f b, v8f c) {
  // D = A(16x4) * B(4x16) + C, fp32, CDNA5 V_WMMA_F32_16X16X4_F32
  return __builtin_amdgcn_wmma_f32_16x16x4_f32(false, a, false, b, (short)0, c,
                                               false, false);
}

// ---------------------------------------------------------------------------
// Kernel 1: x_proj[t*B+b, g] = sum_i enc[b,t,i] * w_ih[g,i] + b_ih[g]
// M=T*B=32768, N=1536, K=1024. Block = 256 thr (8 waves). Each wave: one
// 16-row M tile x four 16-col N tiles. grid = (N/64, M/128).
// ---------------------------------------------------------------------------
__global__ __launch_bounds__(256) void xproj_gemm(
    const float* __restrict__ enc, const float* __restrict__ w_ih,
    const float* __restrict__ b_ih, float* __restrict__ xp) {
  const int lane = threadIdx.x & 31;
  const int wv   = threadIdx.x >> 5;
  const int l15  = lane & 15;
  const int kq   = (lane >> 4) * 2;          // K-pair select (A & B layout)
  const int nb   = blockIdx.x;               // 0..23
  const int mb   = blockIdx.y;               // 0..255

  const int mrow = mb * 128 + wv * 16 + l15; // A row for this lane
  const int t = mrow >> 6, b = mrow & 63;
  const float* arow = enc + ((size_t)b * T_N + t) * U_N + kq;

  const int g0 = nb * 64 + l15;              // B row (= output column) base
  const float* w0 = w_ih + (size_t)(g0 +  0) * U_N + kq;
  const float* w1 = w_ih + (size_t)(g0 + 16) * U_N + kq;
  const float* w2 = w_ih + (size_t)(g0 + 32) * U_N + kq;
  const float* w3 = w_ih + (size_t)(g0 + 48) * U_N + kq;

  v8f acc0 = {}, acc1 = {}, acc2 = {}, acc3 = {};
  for (int k = 0; k < U_N; k += 4) {
    v2f a  = *(const v2f*)(arow + k);
    v2f b0 = *(const v2f*)(w0 + k);
    v2f b1 = *(const v2f*)(w1 + k);
    v2f b2 = *(const v2f*)(w2 + k);
    v2f b3 = *(const v2f*)(w3 + k);
    acc0 = wmma4(a, b0, acc0);
    acc1 = wmma4(a, b1, acc1);
    acc2 = wmma4(a, b2, acc2);
    acc3 = wmma4(a, b3, acc3);
  }
  const float bias0 = b_ih[g0];
  const float bias1 = b_ih[g0 + 16];
  const float bias2 = b_ih[g0 + 32];
  const float bias3 = b_ih[g0 + 48];

  const int mbase = mb * 128 + wv * 16 + ((lane >> 4) * 8);
  for (int e = 0; e < 8; ++e) {
    float* orow = xp + (size_t)(mbase + e) * G_N + nb * 64 + l15;
    orow[0]  = acc0[e] + bias0;
    orow[16] = acc1[e] + bias1;
    orow[32] = acc2[e] + bias2;
    orow[48] = acc3[e] + bias3;
  }
}

// ---------------------------------------------------------------------------
// Kernel 2: persistent bidirectional GRU scan.
// 64 blocks x 128 threads: blocks 0..31 forward, 32..63 backward. Block `cid`
// owns h columns [cid*16, cid*16+16): it computes hp columns
// gate*512+cid*16.. for all 3 gates (48 w_hh rows staged once in ~99 KB LDS)
// and writes its 16 h_new columns; the 32 blocks of a direction then sync via
// a monotonic global-atomic barrier. Wave w (0..3) = M tile (batch rows
// w*16..+15); its three fp32 WMMA accumulators (one per gate) share a single
// A (h_prev) load per k-step; B comes from LDS (ds_load_b64, pad-free banks).
// Per SIMD per step: 3*128 = 384 v_wmma_f32_16x16x4_f32.
// ---------------------------------------------------------------------------
__global__ __launch_bounds__(GRU_THREADS) void gru_scan(
    const float* __restrict__ whh_f, const float* __restrict__ bhh_f,
    const float* __restrict__ whh_b, const float* __restrict__ bhh_b,
    const float* __restrict__ xp_f, const float* __restrict__ xp_b,
    const float* __restrict__ h_init,
    float* __restrict__ h_f, float* __restrict__ h_b,
    unsigned* __restrict__ bars) {
  extern __shared__ float smem[];           // [ROWS][KP] weights + [ROWS] bias

  const int blk = blockIdx.x;
  const int dir = blk >> 5;                 // 0 = fwd, 1 = bwd
  const int cid = blk & 31;                 // h column chunk (16 cols)
  const float* whh = dir ? whh_b : whh_f;
  const float* bhh = dir ? bhh_b : bhh_f;
  const float* xp  = dir ? xp_b  : xp_f;
  float*       hst = dir ? h_b   : h_f;
  unsigned*    bar = bars + dir;

  // Stage the 48 w_hh rows this block needs into LDS (served for 512 steps).
  for (int i = threadIdx.x; i < ROWS * H_N; i += GRU_THREADS) {
    const int c = i >> 9;                   // 0..47
    const int k = i & 511;
    const int grow = (c >> 4) * H_N + cid * CHUNK + (c & 15);
    smem[c * KP + k] = whh[(size_t)grow * H_N + k];
  }
  float* sb = smem + ROWS * KP;
  for (int i = threadIdx.x; i < ROWS; i += GRU_THREADS) {
    const int grow = (i >> 4) * H_N + cid * CHUNK + (i & 15);
    sb[i] = bhh[grow];
  }
  __syncthreads();

  const int lane = threadIdx.x & 31;
  const int mt   = threadIdx.x >> 5;        // wave id == M tile (0..3)
  const int l15  = lane & 15;
  const int kq   = (lane >> 4) * 2;
  const int arow_m = mt * 16 + l15;         // A row (batch index) for lane
  const int cg   = cid * CHUNK + l15;       // column within H
  const int mloc8 = (lane >> 4) * 8;

  const float* bls0 = smem + (0 * CHUNK + l15) * KP + kq;
  const float* bls1 = smem + (1 * CHUNK + l15) * KP + kq;
  const float* bls2 = smem + (2 * CHUNK + l15) * KP + kq;
  const float bb0 = sb[0 * CHUNK + l15];
  const float bb1 = sb[1 * CHUNK + l15];
  const float bb2 = sb[2 * CHUNK + l15];

  for (int s = 0; s < T_N; ++s) {
    const int t_out = dir ? (T_N - 1 - s) : s;
    const int t_prev = dir ? (t_out + 1) : (s - 1);
    const float* hprev =
        (s == 0) ? h_init : hst + (size_t)t_prev * (B_N * H_N);
    const float* ar = hprev + (size_t)arow_m * H_N + kq;

    v8f a0 = {}, a1 = {}, a2 = {};
    for (int k = 0; k < H_N; k += 4) {
      v2f av = *(const v2f*)(ar + k);
      v2f b0 = *(const v2f*)(bls0 + k);
      v2f b1 = *(const v2f*)(bls1 + k);
      v2f b2 = *(const v2f*)(bls2 + k);
      a0 = wmma4(av, b0, a0);
      a1 = wmma4(av, b1, a1);
      a2 = wmma4(av, b2, a2);
    }

    const float* xbase = xp + (size_t)t_out * B_N * G_N;
    for (int e = 0; e < 8; ++e) {
      const int m = mt * 16 + mloc8 + e;    // batch index
      const float* xr = xbase + (size_t)m * G_N;
      const float hp_r = a0[e] + bb0;
      const float hp_z = a1[e] + bb1;
      const float hp_n = a2[e] + bb2;
      const float r = 1.f / (1.f + __expf(-(xr[cg] + hp_r)));
      const float z = 1.f / (1.f + __expf(-(xr[H_N + cg] + hp_z)));
      const float n = tanhf(xr[2 * H_N + cg] + r * hp_n);
      const float hv = hprev[(size_t)m * H_N + cg];
      hst[(size_t)(t_out * B_N + m) * H_N + cg] = (1.f - z) * n + z * hv;
    }

    // Inter-block barrier for this direction (monotonic counter).
    __builtin_amdgcn_fence(__ATOMIC_RELEASE, "agent");  // drain h stores to L2
    __syncthreads();
    if (threadIdx.x == 0) {
      atomicAdd(bar, 1u);
      const unsigned target = (unsigned)NBLK_DIR * (unsigned)(s + 1);
      while (*((volatile unsigned*)bar) < target) {
        __builtin_amdgcn_s_sleep(1);
      }
    }
    __syncthreads();
    __builtin_amdgcn_fence(__ATOMIC_ACQUIRE, "agent");  // invalidate L0/L1
  }
}

// ---------------------------------------------------------------------------
// Kernel 3: emissions[t,b,k] = [h_f ; h_b] . lin_w[k,:] + lin_b[k]
// ---------------------------------------------------------------------------
__global__ __launch_bounds__(256) void emissions_k(
    const float* __restrict__ hf, const float* __restrict__ hb,
    const float* __restrict__ lin_w, const float* __restrict__ lin_b,
    float* __restrict__ em) {
  const int idx = blockIdx.x * 256 + threadIdx.x;
  const int m = idx >> 4;
  const int k = idx & 15;
  if (m >= T_N * B_N || k >= K_N) return;
  const float* h0 = hf + (size_t)m * H_N;
  const float* h1 = hb + (size_t)m * H_N;
  const float* w  = lin_w + (size_t)k * (2 * H_N);
  float acc = lin_b[k];
#pragma unroll 8
  for (int j = 0; j < H_N; ++j) acc += h0[j] * w[j];
#pragma unroll 8
  for (int j = 0; j < H_N; ++j) acc += h1[j] * w[H_N + j];
  em[(size_t)m * K_N + k] = acc;
}

// ---------------------------------------------------------------------------
// Kernel 4: CRF NLL. One block, one thread per batch element (B=64).
// ---------------------------------------------------------------------------
__global__ __launch_bounds__(64) void crf_k(
    const float* __restrict__ em, const int* __restrict__ y,
    const unsigned char* __restrict__ mask, const float* __restrict__ cstart,
    const float* __restrict__ cend, const float* __restrict__ ctrans,
    float* __restrict__ out) {
  __shared__ float st[K_N], se[K_N], tr[K_N * K_N], red[B_N];
  const int b = threadIdx.x;
  if (b < K_N) { st[b] = cstart[b]; se[b] = cend[b]; }
  for (int i = b; i < K_N * K_N; i += B_N) tr[i] = ctrans[i];
  __syncthreads();

  const int* yb = y + (size_t)b * T_N;
  const unsigned char* mb = mask + (size_t)b * T_N;

  float alpha[K_N];
  const int tag0 = yb[0];
  float score = st[tag0] + em[(size_t)b * K_N + tag0];
  for (int k = 0; k < K_N; ++k) alpha[k] = st[k] + em[(size_t)b * K_N + k];
  int prev = tag0;
  int cnt = (int)mb[0];

  for (int t = 1; t < T_N; ++t) {
    const float* e = em + ((size_t)t * B_N + b) * K_N;
    const float mf = (float)mb[t];
    const int tg = yb[t];
    score += mf * (tr[prev * K_N + tg] + e[tg]);
    prev = tg;
    cnt += (int)mb[t];

    float nxt[K_N];
    for (int k = 0; k < K_N; ++k) {
      float mx = -1e30f;
      for (int j = 0; j < K_N; ++j)
        mx = fmaxf(mx, alpha[j] + tr[j * K_N + k]);
      float sm = 0.f;
      for (int j = 0; j < K_N; ++j)
        sm += __expf(alpha[j] + tr[j * K_N + k] - mx);
      nxt[k] = mx + __logf(sm) + e[k];
    }
    if (mb[t]) {
      for (int k = 0; k < K_N; ++k) alpha[k] = nxt[k];
    }
  }
  const int last = yb[cnt - 1];
  score += se[last];

  float mx = -1e30f;
  for (int k = 0; k < K_N; ++k) mx = fmaxf(mx, alpha[k] + se[k]);
  float sm = 0.f;
  for (int k = 0; k < K_N; ++k) sm += __expf(alpha[k] + se[k] - mx);
  const float part = mx + __logf(sm);

  red[b] = score - part;
  __syncthreads();
  for (int off = 32; off > 0; off >>= 1) {
    if (b < off) red[b] += red[b + off];
    __syncthreads();
  }
  if (b == 0) out[0] = -red[0];
}

// ---------------------------------------------------------------------------
// Init: zero barrier counters and h0 every launch (ws is poisoned once).
// ---------------------------------------------------------------------------
__global__ void init_k(float* __restrict__ h_init, unsigned* __restrict__ bars) {
  const int i = blockIdx.x * 256 + threadIdx.x;
  if (i < 2) bars[i] = 0u;
  for (int j = i; j < B_N * H_N; j += gridDim.x * 256) h_init[j] = 0.f;
}

extern "C" void kernel_launch(void* const* d_in, const int* in_sizes, int n_in,
                              void* d_out, int out_size, void* d_ws,
                              size_t ws_size, hipStream_t stream) {
  const float* enc     = (const float*)d_in[0];
  const int*   y_data  = (const int*)d_in[1];
  const unsigned char* masks = (const unsigned char*)d_in[2];
  const float* w_ih_f  = (const float*)d_in[3];
  const float* w_hh_f  = (const float*)d_in[4];
  const float* b_ih_f  = (const float*)d_in[5];
  const float* b_hh_f  = (const float*)d_in[6];
  const float* w_ih_b  = (const float*)d_in[7];
  const float* w_hh_b  = (const float*)d_in[8];
  const float* b_ih_b  = (const float*)d_in[9];
  const float* b_hh_b  = (const float*)d_in[10];
  const float* lin_w   = (const float*)d_in[11];
  const float* lin_b   = (const float*)d_in[12];
  const float* c_start = (const float*)d_in[13];
  const float* c_end   = (const float*)d_in[14];
  const float* c_trans = (const float*)d_in[15];

  float* ws = (float*)d_ws;
  float* xp_f   = ws;                                     // T*B*3H
  float* xp_b   = xp_f + (size_t)T_N * B_N * G_N;         // T*B*3H
  float* h_f    = xp_b + (size_t)T_N * B_N * G_N;         // T*B*H
  float* h_b    = h_f + (size_t)T_N * B_N * H_N;          // T*B*H
  float* h_init = h_b + (size_t)T_N * B_N * H_N;          // B*H
  float* em     = h_init + (size_t)B_N * H_N;             // T*B*K
  unsigned* bars = (unsigned*)(em + (size_t)T_N * B_N * K_N);

  init_k<<<32, 256, 0, stream>>>(h_init, bars);

  dim3 gg(G_N / 64, (T_N * B_N) / 128);
  xproj_gemm<<<gg, 256, 0, stream>>>(enc, w_ih_f, b_ih_f, xp_f);
  xproj_gemm<<<gg, 256, 0, stream>>>(enc, w_ih_b, b_ih_b, xp_b);

  const size_t lds_bytes = (size_t)(ROWS * KP + ROWS) * sizeof(float);
  (void)hipFuncSetAttribute((const void*)gru_scan,
                            hipFuncAttributeMaxDynamicSharedMemorySize,
                            (int)lds_bytes);
  gru_scan<<<2 * NBLK_DIR, GRU_THREADS, lds_bytes, stream>>>(
      w_hh_f, b_hh_f, w_hh_b, b_hh_b, xp_f, xp_b, h_init, h_f, h_b, bars);

  emissions_k<<<(T_N * B_N * 16) / 256, 256, 0, stream>>>(h_f, h_b, lin_w,
                                                          lin_b, em);
  crf_k<<<1, B_N, 0, stream>>>(em, y_data, masks, c_start, c_end, c_trans,
                               (float*)d_out);
}